// KCRouteEncoder_10814727651934
// MI455X (gfx1250) — compile-verified
//
#include <hip/hip_runtime.h>
#include <math.h>

// KCRouteEncoder: per-token 10-way embedding gather + masked softmax + weighted sum.
// Memory-bound: ~19 MB compulsory HBM traffic (idx read + NT out stream),
// ~167 MB L2-resident gather (table is 2.56 MB << 192 MB L2).
// wave32 mapping: one wave per 16 consecutive tokens (looped); lane owns a float2 of the 64-wide embedding
// so each level is exactly one coalesced 256B SADDR-form global_load_b64.

#define NUM_LEVEL 10
#define EMB 64
#define WAVES_PER_BLOCK 8
#define BLOCK_THREADS (WAVES_PER_BLOCK * 32)
#define TOKENS_PER_WAVE 16

typedef float f32x2 __attribute__((ext_vector_type(2)));

__global__ __launch_bounds__(BLOCK_THREADS)
void KCRouteEncoder_kernel(const int* __restrict__ croutes,      // [tokens, 10]
                           const float* __restrict__ emb,        // [10000, 64]
                           const float* __restrict__ rcw,        // [10]
                           float* __restrict__ out,              // [tokens, 64]
                           int tokens) {
    const int lane = threadIdx.x & 31;
    const int wave = blockIdx.x * WAVES_PER_BLOCK + (threadIdx.x >> 5);
    const int t0   = wave * TOKENS_PER_WAVE;
    if (t0 >= tokens) return;

    // Clamped lane for index fetches: all 32 lanes stay active (no exec-mask
    // churn); lanes 10..31 re-read the level-9 word from the same cacheline
    // and are never consumed by readlane.
    const int idx_lane = (lane < NUM_LEVEL) ? lane : (NUM_LEVEL - 1);

    // Level logits are wave-uniform: scalar (s_load) once per wave.
    float w0[NUM_LEVEL];
#pragma unroll
    for (int l = 0; l < NUM_LEVEL; ++l) w0[l] = rcw[l];

    // Software-pipelined index fetch: one 40B coalesced burst per token.
    int my_c_cur = croutes[(size_t)t0 * NUM_LEVEL + idx_lane];

#pragma unroll 1
    for (int ti = 0; ti < TOKENS_PER_WAVE; ++ti) {
        const int token = t0 + ti;
        if (token >= tokens) break;                 // wave-uniform scalar guard

        const int my_c = my_c_cur;

        // Issue next token's index load now (hides behind this token's gathers),
        // and prefetch one token further ahead (gfx1250 global_prefetch_b8).
        if (ti + 1 < TOKENS_PER_WAVE && token + 1 < tokens) {
            my_c_cur = croutes[(size_t)(token + 1) * NUM_LEVEL + idx_lane];
            if (ti + 2 < TOKENS_PER_WAVE && token + 2 < tokens)
                __builtin_prefetch(&croutes[(size_t)(token + 2) * NUM_LEVEL + idx_lane], 0, 1);
        }

        // Broadcast each level's index into SGPRs (v_readlane, compile-time lane):
        // gather base addresses become scalar -> SADDR-form loads, scalar selects.
        int   cl[NUM_LEVEL];
        float wl[NUM_LEVEL];
        float mx = -INFINITY;
#pragma unroll
        for (int l = 0; l < NUM_LEVEL; ++l) {
            cl[l] = __builtin_amdgcn_readlane(my_c, l);
            // available iff related_concept != 0  <=>  croutes != -2
            wl[l] = (cl[l] != -2) ? w0[l] : -INFINITY;
            mx = fmaxf(mx, wl[l]);
        }

        // Softmax over the 10 levels (10 v_exp_f32; trivial vs. memory).
        float a[NUM_LEVEL];
        float ssum = 0.0f;
#pragma unroll
        for (int l = 0; l < NUM_LEVEL; ++l) {
            float e = __expf(wl[l] - mx);
            a[l] = e;
            ssum += e;
        }
        const float rs = 1.0f / ssum;

        // Branchless weighted gather-accumulate; each lane owns e = {2*lane, 2*lane+1}.
        // croutes in {-2,-1} map to zero rows in the reference: clamp row to 0 and
        // zero the coefficient instead of branching (never triggers with valid data).
        f32x2 acc = {0.0f, 0.0f};
#pragma unroll
        for (int l = 0; l < NUM_LEVEL; ++l) {
            const int   row = (cl[l] >= 0) ? cl[l] : 0;          // scalar select
            const float al  = (cl[l] >= 0) ? a[l] * rs : 0.0f;   // zero-row semantics
            const f32x2 v   = ((const f32x2*)(emb + (size_t)row * EMB))[lane];
            acc.x = fmaf(al, v.x, acc.x);
            acc.y = fmaf(al, v.y, acc.y);
        }

        // Streaming output: non-temporal b64 store keeps L2 for the emb table.
        __builtin_nontemporal_store(acc, &((f32x2*)(out + (size_t)token * EMB))[lane]);
    }
}

extern "C" void kernel_launch(void* const* d_in, const int* in_sizes, int n_in,
                              void* d_out, int out_size, void* d_ws, size_t ws_size,
                              hipStream_t stream) {
    const int*   croutes = (const int*)d_in[0];
    // d_in[1] = tailcs: unused by the reference computation.
    const float* emb     = (const float*)d_in[2];
    const float* rcw     = (const float*)d_in[3];
    float*       out     = (float*)d_out;

    const int tokens = in_sizes[1];  // B*S (tailcs element count)
    const int tokens_per_block = WAVES_PER_BLOCK * TOKENS_PER_WAVE;
    const int grid = (tokens + tokens_per_block - 1) / tokens_per_block;

    KCRouteEncoder_kernel<<<grid, BLOCK_THREADS, 0, stream>>>(croutes, emb, rcw, out, tokens);
}